// GraphAttentionLayer_cos_84310208020839
// MI455X (gfx1250) — compile-verified
//
#include <hip/hip_runtime.h>

// GAT layer for MI455X (gfx1250, wave32).
// K1: hT = (X@W)^T via WMMA f32 16x16x4 (accumulators scattered into the
//     64x8192 transpose -- 2 MB, write-once).
// K2: f1/f2 dot products reading hT fully coalesced.
// K3: f2max reduce.
// K4: fused masked-softmax(adj, lrelu(f1+f2^T)) @ h, single pass using the
//     exact unmasked row-max bound m_i = lrelu(f1_i + max_j f2_j) (lrelu is
//     monotone => true row max; all exponents <= 0).
//     K-slot remap: 8 j's per step as two WMMAs with K-sets {j,j+1,j+4,j+5}
//     and {j+2,j+3,j+6,j+7}, so each lane's adj/f2/B-operand traffic is pure
//     B128: 6 vector loads per 8 WMMAs. adj (268 MB, read once, > L2) is
//     loaded non-temporal so it never evicts the L2-resident hT (2 MB).
// HBM floor: 268 MB / 23.3 TB/s ~ 11.5 us; VMEM issue and v_exp_f32 (~67M,
// quarter-rate TRANS over 1024 SIMDs ~ 3 us) now sit below that floor.

typedef float v2f __attribute__((ext_vector_type(2)));
typedef float v8f __attribute__((ext_vector_type(8)));
typedef int   v4i __attribute__((ext_vector_type(4)));

#define NROWS   8192
#define IN_DIM  256
#define OUT_DIM 64
#define SLOPE   0.01f

__device__ __forceinline__ float lrelu(float x) {
    return x > 0.0f ? x : SLOPE * x;
}

#define WMMA_F32(A, B, C) \
    __builtin_amdgcn_wmma_f32_16x16x4_f32(false, (A), false, (B), (short)0, (C), false, false)

// ---------------------------------------------------------------------------
// K1: hT[c][i] = (input @ W)[i][c].  One wave per 16-row tile, 4 N-tiles.
// A (f32 16x4): lanes 0-15 hold M=0..15; VGPR0=K(0|2), VGPR1=K(1|3) by half.
// B (4x16): lane n=lane%16; VGPR0/1 = K rows (0,1)|(2,3) by half.
// ---------------------------------------------------------------------------
__global__ void __launch_bounds__(32)
gemm_h_kernel(const float* __restrict__ input,
              const float* __restrict__ W,
              float* __restrict__ hT) {
    const int lane = threadIdx.x & 31;
    const int r    = lane & 15;
    const int half = lane >> 4;
    const int i0   = blockIdx.x * 16;

    v8f acc0 = {}, acc1 = {}, acc2 = {}, acc3 = {};

    const float* arow = input + (size_t)(i0 + r) * IN_DIM;
    for (int k = 0; k < IN_DIM; k += 4) {
        v2f a;
        a.x = arow[k + 2 * half];
        a.y = arow[k + 2 * half + 1];
        const float* w0 = W + (k + 2 * half) * OUT_DIM;
        const float* w1 = w0 + OUT_DIM;
        v2f b0, b1, b2, b3;
        b0.x = w0[r];      b0.y = w1[r];
        b1.x = w0[16 + r]; b1.y = w1[16 + r];
        b2.x = w0[32 + r]; b2.y = w1[32 + r];
        b3.x = w0[48 + r]; b3.y = w1[48 + r];
        acc0 = WMMA_F32(a, b0, acc0);
        acc1 = WMMA_F32(a, b1, acc1);
        acc2 = WMMA_F32(a, b2, acc2);
        acc3 = WMMA_F32(a, b3, acc3);
    }
    // C/D layout: M = vr + 8*half, N = 16*t + r.  Store transposed.
    for (int vr = 0; vr < 8; ++vr) {
        const int row = i0 + vr + 8 * half;
        hT[(size_t)(r)      * NROWS + row] = acc0[vr];
        hT[(size_t)(16 + r) * NROWS + row] = acc1[vr];
        hT[(size_t)(32 + r) * NROWS + row] = acc2[vr];
        hT[(size_t)(48 + r) * NROWS + row] = acc3[vr];
    }
}

// ---------------------------------------------------------------------------
// K2: f1[i] = h[i,:].a[0:64], f2[i] = h[i,:].a[64:128] -- coalesced hT reads.
// ---------------------------------------------------------------------------
__global__ void __launch_bounds__(256)
f1f2_kernel(const float* __restrict__ hT, const float* __restrict__ a,
            float* __restrict__ f1, float* __restrict__ f2) {
    const int row = blockIdx.x * 256 + threadIdx.x;
    if (row >= NROWS) return;
    float s1 = 0.f, s2 = 0.f;
    for (int c = 0; c < OUT_DIM; ++c) {
        const float v = hT[(size_t)c * NROWS + row];
        s1 += v * a[c];
        s2 += v * a[OUT_DIM + c];
    }
    f1[row] = s1;
    f2[row] = s2;
}

// ---------------------------------------------------------------------------
// K3: global max of f2 (single block, LDS tree reduce).
// ---------------------------------------------------------------------------
__global__ void __launch_bounds__(256)
f2max_kernel(const float* __restrict__ f2, float* __restrict__ f2max) {
    __shared__ float red[256];
    float m = -3.0e38f;
    for (int j = threadIdx.x; j < NROWS; j += 256) m = fmaxf(m, f2[j]);
    red[threadIdx.x] = m;
    __syncthreads();
    for (int s = 128; s > 0; s >>= 1) {
        if (threadIdx.x < s)
            red[threadIdx.x] = fmaxf(red[threadIdx.x], red[threadIdx.x + s]);
        __syncthreads();
    }
    if (threadIdx.x == 0) *f2max = red[0];
}

// ---------------------------------------------------------------------------
// K4: fused masked softmax + alpha@h.
// Block = 8 waves; block owns a 16-row i-tile; wave owns 1/8 of the j-range.
// Per 8-j step: 1x int4 adj (NT) + 1x float4 f2 + 4x float4 hT -> 8 WMMAs.
// ---------------------------------------------------------------------------
__global__ void __launch_bounds__(256)
attn_kernel(const int* __restrict__ adj,
            const float* __restrict__ hT,
            const float* __restrict__ f1,
            const float* __restrict__ f2,
            const float* __restrict__ f2max_p,
            float* __restrict__ out) {
    __shared__ float sC[8 * 16 * OUT_DIM];  // 32 KB: per-wave partial tiles
    __shared__ float sL[8 * 32];            // per-wave per-lane weight sums
    __shared__ float sLrow[16];             // softmax denominators

    const int tid  = threadIdx.x;
    const int wave = tid >> 5;
    const int lane = tid & 31;
    const int r    = lane & 15;
    const int half = lane >> 4;
    const int i0   = blockIdx.x * 16;

    const float f2max = *f2max_p;
    const float f1r   = f1[i0 + r];
    const float mr    = lrelu(f1r + f2max);  // exact unmasked row max

    const int jbeg = wave * (NROWS / 8);
    const int jend = jbeg + (NROWS / 8);

    const int*   adjrow = adj + (size_t)(i0 + r) * NROWS;
    const float* hT0 = hT + (size_t)(r)      * NROWS;
    const float* hT1 = hT + (size_t)(16 + r) * NROWS;
    const float* hT2 = hT + (size_t)(32 + r) * NROWS;
    const float* hT3 = hT + (size_t)(48 + r) * NROWS;

    v8f acc0 = {}, acc1 = {}, acc2 = {}, acc3 = {};
    float lsum = 0.f;

    for (int j = jbeg; j < jend; j += 8) {
        const int jb = j + 4 * half;  // this lane's 4 contiguous columns
        // Streaming adj: non-temporal (read-once, bigger than L2) + prefetch
        // ahead on this lane's row to hide the 16-way row-strided pattern.
        const v4i aa = __builtin_nontemporal_load(
                           reinterpret_cast<const v4i*>(adjrow + jb));
        __builtin_prefetch(adjrow + jb + 1024, 0, 0);
        const float4 ff = *reinterpret_cast<const float4*>(f2 + jb);

        const float w0 = (aa.x > 0) ? __expf(lrelu(f1r + ff.x) - mr) : 0.0f;
        const float w1 = (aa.y > 0) ? __expf(lrelu(f1r + ff.y) - mr) : 0.0f;
        const float w2 = (aa.z > 0) ? __expf(lrelu(f1r + ff.z) - mr) : 0.0f;
        const float w3 = (aa.w > 0) ? __expf(lrelu(f1r + ff.w) - mr) : 0.0f;
        lsum += (w0 + w1) + (w2 + w3);

        // K-slot remap: wmma A covers K-set {j,j+1,j+4,j+5}: a0 = (w0,w1);
        // wmma B covers K-set {j+2,j+3,j+6,j+7}: a1 = (w2,w3).
        v2f a0; a0.x = w0; a0.y = w1;
        v2f a1; a1.x = w2; a1.y = w3;

        const float4 hv0 = *reinterpret_cast<const float4*>(hT0 + jb);
        const float4 hv1 = *reinterpret_cast<const float4*>(hT1 + jb);
        const float4 hv2 = *reinterpret_cast<const float4*>(hT2 + jb);
        const float4 hv3 = *reinterpret_cast<const float4*>(hT3 + jb);

        v2f b;
        b.x = hv0.x; b.y = hv0.y; acc0 = WMMA_F32(a0, b, acc0);
        b.x = hv1.x; b.y = hv1.y; acc1 = WMMA_F32(a0, b, acc1);
        b.x = hv2.x; b.y = hv2.y; acc2 = WMMA_F32(a0, b, acc2);
        b.x = hv3.x; b.y = hv3.y; acc3 = WMMA_F32(a0, b, acc3);
        b.x = hv0.z; b.y = hv0.w; acc0 = WMMA_F32(a1, b, acc0);
        b.x = hv1.z; b.y = hv1.w; acc1 = WMMA_F32(a1, b, acc1);
        b.x = hv2.z; b.y = hv2.w; acc2 = WMMA_F32(a1, b, acc2);
        b.x = hv3.z; b.y = hv3.w; acc3 = WMMA_F32(a1, b, acc3);
    }

    // Dump per-wave partials (C/D layout: M = vr + 8*half, N = 16*t + r).
    for (int vr = 0; vr < 8; ++vr) {
        const int M = vr + 8 * half;
        float* dst = sC + (wave * 16 + M) * OUT_DIM;
        dst[r]      = acc0[vr];
        dst[16 + r] = acc1[vr];
        dst[32 + r] = acc2[vr];
        dst[48 + r] = acc3[vr];
    }
    sL[wave * 32 + lane] = lsum;
    __syncthreads();

    // Softmax denominators: lanes r and r+16 of every wave hold row-r partials.
    if (tid < 16) {
        float l = 0.f;
        for (int w = 0; w < 8; ++w)
            l += sL[w * 32 + tid] + sL[w * 32 + 16 + tid];
        sLrow[tid] = l;
    }
    __syncthreads();

    // Cross-wave reduce of the 16x64 tile + normalize + store.
    for (int idx = tid; idx < 16 * OUT_DIM; idx += 256) {
        float s = 0.f;
        for (int w = 0; w < 8; ++w) s += sC[w * (16 * OUT_DIM) + idx];
        const int M = idx >> 6;
        out[(size_t)(i0 + M) * OUT_DIM + (idx & 63)] = s / sLrow[M];
    }

    // Scalar outputs: e[1,2], e[1,3], e[3,2].
    if (blockIdx.x == 0 && tid == 0) {
        out[NROWS * OUT_DIM + 0] = lrelu(f1[1] + f2[2]);
        out[NROWS * OUT_DIM + 1] = lrelu(f1[1] + f2[3]);
        out[NROWS * OUT_DIM + 2] = lrelu(f1[3] + f2[2]);
    }
}

// ---------------------------------------------------------------------------
extern "C" void kernel_launch(void* const* d_in, const int* in_sizes, int n_in,
                              void* d_out, int out_size, void* d_ws, size_t ws_size,
                              hipStream_t stream) {
    (void)in_sizes; (void)n_in; (void)out_size; (void)ws_size;
    const float* input = (const float*)d_in[0];
    const int*   adj   = (const int*)d_in[1];
    const float* W     = (const float*)d_in[2];
    const float* a     = (const float*)d_in[3];
    float* out = (float*)d_out;

    // Workspace: hT (2 MB) | f1 (32 KB) | f2 (32 KB) | f2max (4 B)
    float* hT    = (float*)d_ws;
    float* f1    = hT + (size_t)OUT_DIM * NROWS;
    float* f2    = f1 + NROWS;
    float* f2max = f2 + NROWS;

    gemm_h_kernel<<<NROWS / 16, 32, 0, stream>>>(input, W, hT);
    f1f2_kernel<<<NROWS / 256, 256, 0, stream>>>(hT, a, f1, f2);
    f2max_kernel<<<1, 256, 0, stream>>>(f2, f2max);
    attn_kernel<<<NROWS / 16, 256, 0, stream>>>(adj, hT, f1, f2, f2max, out);
}